// GAT_41231686042229
// MI455X (gfx1250) — compile-verified
//
#include <hip/hip_runtime.h>
#include <hip/hip_bf16.h>

// GAT on MI455X (gfx1250). GEMMs via V_WMMA_F32_16X16X4_F32 (fp32 matrix core path,
// NT column-tile blocking for A-fragment reuse), edge softmax via ordered-uint atomic
// max + fully-coalesced wave-per-edge f32 atomic accumulation, LN via wave32 shuffles.

#define HID 32
#define HEADS 8
#define OUT_DIM 16
#define NEG_SLOPE 0.2f
#define LN_EPS 1e-5f

typedef __attribute__((ext_vector_type(2))) float v2f;
typedef __attribute__((ext_vector_type(8))) float v8f;

// ---------- helpers ----------
__device__ __forceinline__ unsigned f2ord(float f) {
    unsigned u = __float_as_uint(f);
    return (u & 0x80000000u) ? ~u : (u | 0x80000000u);
}
__device__ __forceinline__ float ord2f(unsigned o) {
    unsigned u = (o & 0x80000000u) ? (o & 0x7FFFFFFFu) : ~o;
    return __uint_as_float(u);
}
#define ORD_NEG_INF 0x007FFFFFu  // f2ord(-inf)

__device__ __forceinline__ void edge_sd(const int* ei, int e, int E_, int& s, int& d) {
    if (e < E_) { s = ei[e]; d = ei[E_ + e]; }
    else        { s = e - E_; d = s; }          // self loops appended after edges
}

// ---------- fills ----------
__global__ void fill_u32_k(unsigned* p, unsigned v, long n) {
    long i = (long)blockIdx.x * blockDim.x + threadIdx.x;
    if (i < n) p[i] = v;
}
__global__ void fill_f32_k(float* p, float v, long n) {
    long i = (long)blockIdx.x * blockDim.x + threadIdx.x;
    if (i < n) p[i] = v;
}

// ---------- input layer: h0[n,j] = x[n]*Win[j] + bin[j] ----------
__global__ void in_layer_k(const float* __restrict__ x, const float* __restrict__ Win,
                           const float* __restrict__ bin, float* __restrict__ h0, int Nn) {
    long t = (long)blockIdx.x * blockDim.x + threadIdx.x;
    if (t >= (long)Nn * HID) return;
    int j = (int)(t & (HID - 1));
    int n = (int)(t >> 5);
    h0[t] = x[n] * Win[j] + bin[j];
}

// ---------- dual GEMM with fp32 WMMA, NT column tiles per wave ----------
// A: [M x K] row-major. B1/B2: [K x C] row-major. out1 = A@B1 + bias1, out2 = A@B2 (+bias2).
// grid = (M/16, C/(16*NT)), block = 32 (one wave32). A fragment reused across 2*NT WMMAs/k-step.
template<int NT>
__global__ void gemm_dual_wmma_k(const float* __restrict__ A,
                                 const float* __restrict__ B1, const float* __restrict__ bias1,
                                 float* __restrict__ out1,
                                 const float* __restrict__ B2, const float* __restrict__ bias2,
                                 float* __restrict__ out2,
                                 int M, int K, int C) {
    const int lane  = threadIdx.x & 31;
    const int m0    = blockIdx.x * 16;
    const int n0    = blockIdx.y * (16 * NT);
    const int mrow  = lane & 15;          // A: lanes 0-15 and 16-31 both carry M=0..15
    const int nb    = lane & 15;          // B/C: N = lane % 16
    const int khalf = (lane >> 4) * 2;    // A/B VGPR0 holds K = khalf, VGPR1 holds K = khalf+1

    v8f zero = {};
    v8f c1[NT], c2[NT];
#pragma unroll
    for (int t = 0; t < NT; ++t) { c1[t] = zero; c2[t] = zero; }

    const float* arow = A + (long)(m0 + mrow) * K;

    for (int k0 = 0; k0 < K; k0 += 4) {
        const int k = k0 + khalf;
        v2f a;
        a.x = arow[k];                    // consecutive pair -> single b64 load
        a.y = arow[k + 1];
#pragma unroll
        for (int t = 0; t < NT; ++t) {
            const int n = n0 + t * 16 + nb;
            v2f b1, b2;
            b1.x = B1[(long)k * C + n];
            b1.y = B1[(long)(k + 1) * C + n];
            b2.x = B2[(long)k * C + n];
            b2.y = B2[(long)(k + 1) * C + n];
            // 8-arg form: (neg_a, A, neg_b, B, c_mod, C, reuse_a, reuse_b)
            c1[t] = __builtin_amdgcn_wmma_f32_16x16x4_f32(false, a, false, b1, (short)0, c1[t], false, false);
            c2[t] = __builtin_amdgcn_wmma_f32_16x16x4_f32(false, a, false, b2, (short)0, c2[t], false, false);
        }
    }

    // C/D layout: VGPR v -> M = v + 8*(lane/16), N = lane%16
    const int mbase = m0 + 8 * (lane >> 4);
#pragma unroll
    for (int t = 0; t < NT; ++t) {
        const int n = n0 + t * 16 + nb;
        const float bb1 = bias1 ? bias1[n] : 0.0f;
        const float bb2 = bias2 ? bias2[n] : 0.0f;
#pragma unroll
        for (int v = 0; v < 8; ++v) {
            out1[(long)(mbase + v) * C + n] = c1[t][v] + bb1;
            out2[(long)(mbase + v) * C + n] = c2[t][v] + bb2;
        }
    }
}

// ---------- attention scores: a_s[n,h] = <gh[n,h,:], att_s[h,:]>, same for a_d ----------
__global__ void att_scores_k(const float* __restrict__ gh, const float* __restrict__ att_s,
                             const float* __restrict__ att_d, float* __restrict__ a_s,
                             float* __restrict__ a_d, int Nn, int H) {
    long t = (long)blockIdx.x * blockDim.x + threadIdx.x;
    if (t >= (long)Nn * H) return;
    int h = (int)(t % H);
    long n = t / H;
    const float* row = gh + (n * H + h) * (long)HID;
    float s = 0.f, dd = 0.f;
#pragma unroll
    for (int f = 0; f < HID; ++f) {
        float v = row[f];
        s  += v * att_s[h * HID + f];
        dd += v * att_d[h * HID + f];
    }
    a_s[t] = s;
    a_d[t] = dd;
}

// ---------- edge pass 1: segment max (ordered-uint atomic max) ----------
__global__ void edge_max_k(const int* __restrict__ ei, const float* __restrict__ a_s,
                           const float* __restrict__ a_d, unsigned* __restrict__ emax,
                           int E_, int Nn, int H) {
    long t = (long)blockIdx.x * blockDim.x + threadIdx.x;
    long total = (long)(E_ + Nn) * H;
    if (t >= total) return;
    int h = (int)(t % H);
    int e = (int)(t / H);
    int s, d;
    edge_sd(ei, e, E_, s, d);
    float v = a_s[(long)s * H + h] + a_d[(long)d * H + h];
    v = v > 0.f ? v : v * NEG_SLOPE;
    atomicMax(&emax[(long)d * H + h], f2ord(v));
}

// ---------- edge pass 2: segment sum of exp(e - max) ----------
__global__ void edge_expsum_k(const int* __restrict__ ei, const float* __restrict__ a_s,
                              const float* __restrict__ a_d, const unsigned* __restrict__ emax,
                              float* __restrict__ denom, int E_, int Nn, int H) {
    long t = (long)blockIdx.x * blockDim.x + threadIdx.x;
    long total = (long)(E_ + Nn) * H;
    if (t >= total) return;
    int h = (int)(t % H);
    int e = (int)(t / H);
    int s, d;
    edge_sd(ei, e, E_, s, d);
    float v = a_s[(long)s * H + h] + a_d[(long)d * H + h];
    v = v > 0.f ? v : v * NEG_SLOPE;
    float ex = __expf(v - ord2f(emax[(long)d * H + h]));
    atomicAdd(&denom[(long)d * H + h], ex);
}

// ---------- edge pass 3: acc[dst,h,:] += alpha * gh[src,h,:] ----------
// One wave32 per edge; lane = feature. All loads and atomic adds are coalesced
// 128B cacheline transactions (this phase carries ~870MB of gather traffic and
// is the roofline-dominant part of the whole network).
template<int H>
__global__ void edge_acc_wave_k(const int* __restrict__ ei, const float* __restrict__ a_s,
                                const float* __restrict__ a_d, const unsigned* __restrict__ emax,
                                const float* __restrict__ denom, const float* __restrict__ gh,
                                float* __restrict__ acc, int E_, int Nn) {
    long wave = ((long)blockIdx.x * blockDim.x + threadIdx.x) >> 5;
    int lane = threadIdx.x & 31;
    long EP = (long)E_ + Nn;
    if (wave >= EP) return;
    int s, d;
    edge_sd(ei, (int)wave, E_, s, d);
    const float* src_base = gh + (long)s * H * HID;
    float*       dst_base = acc + (long)d * H * HID;
#pragma unroll
    for (int h = 0; h < H; ++h) {
        long sh = (long)s * H + h, dh = (long)d * H + h;
        float v = a_s[sh] + a_d[dh];                 // lane-uniform broadcast loads
        v = v > 0.f ? v : v * NEG_SLOPE;
        float alpha = __expf(v - ord2f(emax[dh])) / (denom[dh] + 1e-16f);
        atomicAdd(&dst_base[h * HID + lane], src_base[h * HID + lane] * alpha);
    }
}

// ---------- (ELU) + residual + LayerNorm, one wave32 per node ----------
// out may alias acc (each thread reads all its elements before writing).
__global__ void elu_res_ln_k(const float* __restrict__ acc, const float* __restrict__ gbias,
                             const float* __restrict__ res, const float* __restrict__ g,
                             const float* __restrict__ b, float* __restrict__ out,
                             int Nn, int D, int applyElu) {
    int wave = (int)(((long)blockIdx.x * blockDim.x + threadIdx.x) >> 5);
    int lane = threadIdx.x & 31;
    if (wave >= Nn) return;
    const int per = D >> 5;   // D in {32, 256}
    float vals[8];
    float sum = 0.f;
    for (int i = 0; i < per; ++i) {
        int j = i * 32 + lane;
        float v = acc[(long)wave * D + j] + gbias[j];
        if (applyElu) v = v > 0.f ? v : (__expf(v) - 1.0f);
        v += res[(long)wave * D + j];
        vals[i] = v;
        sum += v;
    }
    for (int o = 16; o > 0; o >>= 1) sum += __shfl_xor(sum, o, 32);
    float mean = sum / (float)D;
    float var = 0.f;
    for (int i = 0; i < per; ++i) {
        float dv = vals[i] - mean;
        var += dv * dv;
    }
    for (int o = 16; o > 0; o >>= 1) var += __shfl_xor(var, o, 32);
    float rstd = rsqrtf(var / (float)D + LN_EPS);
    for (int i = 0; i < per; ++i) {
        int j = i * 32 + lane;
        out[(long)wave * D + j] = (vals[i] - mean) * rstd * g[j] + b[j];
    }
}

// ---------- global mean pool: partial sums in LDS, one atomic per column per block ----------
__global__ void pool_sum_k(const float* __restrict__ h, float* __restrict__ pooled, int Nn) {
    __shared__ float sh[HID];
    if (threadIdx.x < HID) sh[threadIdx.x] = 0.f;
    __syncthreads();
    long t = (long)blockIdx.x * blockDim.x + threadIdx.x;
    if (t < (long)Nn * HID) atomicAdd(&sh[t & (HID - 1)], h[t]);
    __syncthreads();
    if (threadIdx.x < HID) atomicAdd(&pooled[threadIdx.x], sh[threadIdx.x]);
}

// ---------- final projection ----------
__global__ void out_proj_k(const float* __restrict__ pooled, const float* __restrict__ Wout,
                           const float* __restrict__ bout, float* __restrict__ out, float invN) {
    int j = threadIdx.x;
    if (j >= OUT_DIM) return;
    float s = bout[j];
#pragma unroll
    for (int k = 0; k < HID; ++k)
        s += pooled[k] * invN * Wout[k * OUT_DIM + j];
    out[j] = s;
}

// =======================================================================================
extern "C" void kernel_launch(void* const* d_in, const int* in_sizes, int n_in,
                              void* d_out, int out_size, void* d_ws, size_t ws_size,
                              hipStream_t stream) {
    const float* x    = (const float*)d_in[0];
    const int*   ei   = (const int*)d_in[1];    // [2,E] flattened: row0 = src, row1 = dst
    // d_in[2] = batch (all zeros, single graph) -> unused
    const float* Win  = (const float*)d_in[3];
    const float* bin_ = (const float*)d_in[4];
    const float* Wr0  = (const float*)d_in[5];
    const float* br0  = (const float*)d_in[6];
    const float* W0   = (const float*)d_in[7];
    const float* as0  = (const float*)d_in[8];
    const float* ad0  = (const float*)d_in[9];
    const float* bg0  = (const float*)d_in[10];
    const float* g0   = (const float*)d_in[11];
    const float* b0   = (const float*)d_in[12];
    const float* Wr1  = (const float*)d_in[13];
    const float* br1  = (const float*)d_in[14];
    const float* W1   = (const float*)d_in[15];
    const float* as1  = (const float*)d_in[16];
    const float* ad1  = (const float*)d_in[17];
    const float* bg1  = (const float*)d_in[18];
    const float* g1   = (const float*)d_in[19];
    const float* b1   = (const float*)d_in[20];
    const float* Wout = (const float*)d_in[21];
    const float* bout = (const float*)d_in[22];
    float* out = (float*)d_out;

    const int Nn = in_sizes[0];         // 50000 (multiple of 16)
    const int E  = in_sizes[1] / 2;     // 800000
    const int D0 = HID * HEADS;         // 256

    // ---- workspace carve (peak ~ N*832 floats ~ 167 MB) ----
    float* ws = (float*)d_ws;
    long o = 0;
    float*    h0    = ws + o; o += (long)Nn * HID;   // [N,32]
    float*    res0  = ws + o; o += (long)Nn * D0;    // [N,256]  (reused for layer-1 buffers)
    float*    gh0   = ws + o; o += (long)Nn * D0;    // [N,256]
    float*    as0v  = ws + o; o += (long)Nn * HEADS; // [N,8]
    float*    ad0v  = ws + o; o += (long)Nn * HEADS;
    unsigned* emax0 = (unsigned*)(ws + o); o += (long)Nn * HEADS;
    float*    den0  = ws + o; o += (long)Nn * HEADS;
    float*    acc0  = ws + o; o += (long)Nn * D0;    // [N,256]; LN layer0 runs in place -> hL0
    float*    pooled= ws + o; o += 64;

    // layer-1 aliases over dead layer-0 buffers
    float*    res1  = res0;                            // [N,32]
    float*    gh1   = res0 + (long)Nn * HID;           // [N,32]
    float*    acc1  = res0 + (long)Nn * HID * 2;       // [N,32]; LN layer1 in place -> hL1
    float*    as1v  = as0v;
    float*    ad1v  = ad0v;
    unsigned* emax1 = emax0;
    float*    den1  = den0;
    float*    hL0   = acc0;
    float*    hL1   = acc1;

    const int B = 256;
    const long EP = (long)(E + Nn);                    // edges + self loops

    // ===== input layer =====
    in_layer_k<<<(int)(((long)Nn * HID + B - 1) / B), B, 0, stream>>>(x, Win, bin_, h0, Nn);

    // ===== layer 0 =====
    // res0 = h0@Wr0 + br0 ; gh0 = h0@W0   (WMMA; 4 column tiles x 2 weights per wave)
    gemm_dual_wmma_k<4><<<dim3(Nn / 16, D0 / 64), 32, 0, stream>>>(
        h0, Wr0, br0, res0, W0, nullptr, gh0, Nn, HID, D0);

    att_scores_k<<<(int)(((long)Nn * HEADS + B - 1) / B), B, 0, stream>>>(
        gh0, as0, ad0, as0v, ad0v, Nn, HEADS);

    fill_u32_k<<<(int)(((long)Nn * HEADS + B - 1) / B), B, 0, stream>>>(emax0, ORD_NEG_INF, (long)Nn * HEADS);
    fill_f32_k<<<(int)(((long)Nn * HEADS + B - 1) / B), B, 0, stream>>>(den0, 0.f, (long)Nn * HEADS);
    fill_f32_k<<<(int)(((long)Nn * D0 + B - 1) / B), B, 0, stream>>>(acc0, 0.f, (long)Nn * D0);

    {
        long tot = EP * HEADS;
        int blocks = (int)((tot + B - 1) / B);
        edge_max_k   <<<blocks, B, 0, stream>>>(ei, as0v, ad0v, emax0, E, Nn, HEADS);
        edge_expsum_k<<<blocks, B, 0, stream>>>(ei, as0v, ad0v, emax0, den0, E, Nn, HEADS);
        // wave-per-edge accumulation: blocks of 8 waves
        edge_acc_wave_k<HEADS><<<(int)((EP + 7) / 8), B, 0, stream>>>(
            ei, as0v, ad0v, emax0, den0, gh0, acc0, E, Nn);
    }

    // hL0 = LN(elu(acc0 + bg0) + res0) * g0 + b0   (in place over acc0)
    elu_res_ln_k<<<(Nn * 32 + B - 1) / B, B, 0, stream>>>(acc0, bg0, res0, g0, b0, hL0, Nn, D0, 1);

    // ===== layer 1 =====
    // res1 = hL0@Wr1 + br1 ; gh1 = hL0@W1   (res0 region is dead now; outputs alias it)
    gemm_dual_wmma_k<2><<<dim3(Nn / 16, 1), 32, 0, stream>>>(
        hL0, Wr1, br1, res1, W1, nullptr, gh1, Nn, D0, HID);

    att_scores_k<<<(int)(((long)Nn + B - 1) / B), B, 0, stream>>>(gh1, as1, ad1, as1v, ad1v, Nn, 1);

    fill_u32_k<<<(Nn + B - 1) / B, B, 0, stream>>>(emax1, ORD_NEG_INF, Nn);
    fill_f32_k<<<(Nn + B - 1) / B, B, 0, stream>>>(den1, 0.f, Nn);
    fill_f32_k<<<(int)(((long)Nn * HID + B - 1) / B), B, 0, stream>>>(acc1, 0.f, (long)Nn * HID);
    fill_f32_k<<<1, 64, 0, stream>>>(pooled, 0.f, 64);

    {
        long tot = EP;
        int blocks = (int)((tot + B - 1) / B);
        edge_max_k   <<<blocks, B, 0, stream>>>(ei, as1v, ad1v, emax1, E, Nn, 1);
        edge_expsum_k<<<blocks, B, 0, stream>>>(ei, as1v, ad1v, emax1, den1, E, Nn, 1);
        edge_acc_wave_k<1><<<(int)((EP + 7) / 8), B, 0, stream>>>(
            ei, as1v, ad1v, emax1, den1, gh1, acc1, E, Nn);
    }

    // hL1 = LN((acc1 + bg1) + res1) * g1 + b1   (no ELU on last layer; in place over acc1)
    elu_res_ln_k<<<(Nn * 32 + B - 1) / B, B, 0, stream>>>(acc1, bg1, res1, g1, b1, hL1, Nn, HID, 0);

    // ===== pool + output =====
    pool_sum_k<<<(int)(((long)Nn * HID + B - 1) / B), B, 0, stream>>>(hL1, pooled, Nn);
    out_proj_k<<<1, 32, 0, stream>>>(pooled, Wout, bout, out, 1.0f / (float)Nn);
}